// LogPolar_2851858285176
// MI455X (gfx1250) — compile-verified
//
#include <hip/hip_runtime.h>
#include <math.h>

// Log-polar bilinear remap, (128, 384, 384, 3) f32 -> same shape.
// Memory-bound gather: ~450 MB total traffic -> ~19.5 us floor at 23.3 TB/s.
// Per-(i,j) sampling math hoisted out of the batch loop; NT stores keep L2
// for the gathered input; global_prefetch_b8 hides next-batch gather latency.

namespace {

constexpr int   kN     = 384;            // rows (theta)
constexpr int   kM     = 384;            // cols (rho)
constexpr int   kD     = 3;              // channels
constexpr int   kB     = 128;            // batch
constexpr int   kPlane = kN * kM * kD;   // elements per image = 442368
constexpr int   kBPC   = 8;              // batches per chunk (per block)

struct __attribute__((aligned(4))) F3 { float x, y, z; };

} // namespace

extern "C" __global__ void __launch_bounds__(256)
logpolar_remap_kernel(const float* __restrict__ in, float* __restrict__ out)
{
    const int pix = blockIdx.x * 256 + threadIdx.x;
    if (pix >= kN * kM) return;

    const int i = pix / kM;        // theta index (row)
    const int j = pix - i * kM;    // rho index (col)

    // theta = 2*pi * i / 384 ; rho = exp(j/384 * ln(192)) ; center (192,192)
    const float theta = 0.016362461737446838f * (float)i;   // 2*pi/384
    const float rho   = __expf(0.013691394197989013f * (float)j); // ln(192)/384
    float sn, cs;
    __sincosf(theta, &sn, &cs);
    const float xs = 192.0f + rho * cs;
    const float ys = 192.0f + rho * sn;

    const float xf = floorf(xs), yf = floorf(ys);
    const int   x0 = (int)xf,    y0 = (int)yf;
    const int   x1 = x0 + 1,     y1 = y0 + 1;
    const float wx = xs - xf,    wy = ys - yf;

    // Out-of-bounds masks folded directly into the bilinear weights.
    const float mx0 = (x0 >= 0 && x0 < kM) ? 1.0f : 0.0f;
    const float mx1 = (x1 >= 0 && x1 < kM) ? 1.0f : 0.0f;
    const float my0 = (y0 >= 0 && y0 < kN) ? 1.0f : 0.0f;
    const float my1 = (y1 >= 0 && y1 < kN) ? 1.0f : 0.0f;

    const float w00 = (1.0f - wy) * (1.0f - wx) * my0 * mx0;
    const float w01 = (1.0f - wy) * wx          * my0 * mx1;
    const float w10 = wy          * (1.0f - wx) * my1 * mx0;
    const float w11 = wy          * wx          * my1 * mx1;

    const int x0c = min(max(x0, 0), kM - 1);
    const int x1c = min(max(x1, 0), kM - 1);
    const int y0c = min(max(y0, 0), kN - 1);
    const int y1c = min(max(y1, 0), kN - 1);

    // Element offsets of the 4 gather corners within one image plane.
    const int o00 = (y0c * kM + x0c) * kD;
    const int o01 = (y0c * kM + x1c) * kD;
    const int o10 = (y1c * kM + x0c) * kD;
    const int o11 = (y1c * kM + x1c) * kD;

    const int    b0 = blockIdx.y * kBPC;
    const float* pb = in  + (size_t)b0 * kPlane;
    float*       ob = out + (size_t)b0 * kPlane + (size_t)pix * kD;

#pragma unroll
    for (int b = 0; b < kBPC; ++b) {
        // Prefetch next batch's gather rows (addresses are affine in b).
        // Speculative: safe even past the end of the buffer on gfx1250.
        __builtin_prefetch(pb + kPlane + o00, 0, 1);
        __builtin_prefetch(pb + kPlane + o10, 0, 1);

        const F3 v00 = *(const F3*)(pb + o00);
        const F3 v01 = *(const F3*)(pb + o01);
        const F3 v10 = *(const F3*)(pb + o10);
        const F3 v11 = *(const F3*)(pb + o11);

        const float r0 = v00.x * w00 + v01.x * w01 + v10.x * w10 + v11.x * w11;
        const float r1 = v00.y * w00 + v01.y * w01 + v10.y * w10 + v11.y * w11;
        const float r2 = v00.z * w00 + v01.z * w01 + v10.z * w10 + v11.z * w11;

        // Write-once output: non-temporal so L2 stays full of gathered input.
        __builtin_nontemporal_store(r0, ob + 0);
        __builtin_nontemporal_store(r1, ob + 1);
        __builtin_nontemporal_store(r2, ob + 2);

        pb += kPlane;
        ob += kPlane;
    }
}

extern "C" void kernel_launch(void* const* d_in, const int* in_sizes, int n_in,
                              void* d_out, int out_size, void* d_ws, size_t ws_size,
                              hipStream_t stream)
{
    (void)in_sizes; (void)n_in; (void)out_size; (void)d_ws; (void)ws_size;

    const float* x   = (const float*)d_in[0];
    float*       out = (float*)d_out;

    // grid.x: pixel blocks (fast dim) ; grid.y: batch chunks (slow dim) so all
    // pixel-blocks of one 8-image chunk (~14 MB) run together -> L2 resident.
    dim3 grid((kN * kM + 255) / 256, kB / kBPC, 1);
    logpolar_remap_kernel<<<grid, 256, 0, stream>>>(x, out);
}